// SemanticEncoder_10677288698548
// MI455X (gfx1250) — compile-verified
//
#include <hip/hip_runtime.h>
#include <stdint.h>

// N=262144 rows, L=5 steps, H=128 channels (64 complex pairs).
// out[n,h] = mean_l rotate(e[n,l,:], rot[l])  -- pure streaming, HBM-bound.
//
// CDNA5 path: per-wave double-buffered global_load_async_to_lds_b128 pipeline
// (ASYNCcnt), ds_load_b128 consumption, NT cache hints on the 671MB stream,
// nontemporal b128 stores for the output.

typedef float v4f __attribute__((ext_vector_type(4)));

#define LSTEPS 5
#define HID    128
#define HPAIR  64
#define WAVES_PER_BLK 8
#define THREADS (WAVES_PER_BLK * 32)

__device__ __forceinline__ void async_row_load(const float* __restrict__ gbase,
                                               const v4f* lbase, int lane) {
  // One row slab: L*H floats = 2560B; wave loads it as 5 x (32 lanes x 16B).
#pragma unroll
  for (int l = 0; l < LSTEPS; ++l) {
    const float* g = gbase + l * HID + lane * 4;
    unsigned lds = (unsigned)reinterpret_cast<unsigned long long>(lbase + l * 32 + lane);
    asm volatile("global_load_async_to_lds_b128 %0, %1, off th:TH_LOAD_NT"
                 :: "v"(lds), "v"(g) : "memory");
  }
}

__global__ __launch_bounds__(THREADS)
void semenc_rotmean_kernel(const float* __restrict__ edata,
                           const float* __restrict__ rvec,
                           float* __restrict__ out, int nrows) {
  __shared__ __align__(16) float rot_s[LSTEPS * HID];          // 2560 B: [l][pair][2]
  __shared__ v4f sbuf[WAVES_PER_BLK * 2 * LSTEPS * 32];        // 40960 B: per-wave double buffer

  const int tid  = threadIdx.x;
  const int lane = tid & 31;
  const int wv   = tid >> 5;

  // ---- build cumulative rotation table (threads 0..63, one channel pair each) ----
  if (tid < HPAIR) {
    const int k = tid;
    float rvx[4], rvy[4];
#pragma unroll
    for (int r = 0; r < 4; ++r) {
      float x = rvec[r * HID + k * 2 + 0];
      float y = rvec[r * HID + k * 2 + 1];
      float inv = 1.0f / fmaxf(sqrtf(x * x + y * y), 1e-12f);
      rvx[r] = x * inv; rvy[r] = y * inv;
    }
    // ETYPES=(0,2,5,7) into conj-expanded table: T[i] multiplied at step i
    float tx[4] = { rvx[0], rvx[2],  rvx[1],  rvx[3] };
    float ty[4] = { rvy[0], rvy[2], -rvy[1], -rvy[3] };
    float cx = 1.0f, cy = 0.0f;                       // rot[L-1] = identity
    rot_s[(LSTEPS - 1) * HID + k * 2 + 0] = cx;
    rot_s[(LSTEPS - 1) * HID + k * 2 + 1] = cy;
#pragma unroll
    for (int i = LSTEPS - 2; i >= 0; --i) {
      float re = cx * tx[i] - cy * ty[i];
      float im = cx * ty[i] + cy * tx[i];
      cx = re; cy = im;
      rot_s[i * HID + k * 2 + 0] = cx;
      rot_s[i * HID + k * 2 + 1] = cy;
    }
  }
  __syncthreads();

  // Each lane owns pairs (2*lane, 2*lane+1): grab its 5 rotation float4s.
  v4f rt[LSTEPS];
#pragma unroll
  for (int l = 0; l < LSTEPS; ++l)
    rt[l] = *reinterpret_cast<const v4f*>(&rot_s[l * HID + lane * 4]);

  const int gwave  = blockIdx.x * WAVES_PER_BLK + wv;
  const int nwaves = gridDim.x * WAVES_PER_BLK;
  const v4f* buf0 = sbuf + (wv * 2 + 0) * LSTEPS * 32;
  const v4f* buf1 = sbuf + (wv * 2 + 1) * LSTEPS * 32;

  if (gwave < nrows)
    async_row_load(edata + (size_t)gwave * (LSTEPS * HID), buf0, lane);

  for (int it = 0; ; ++it) {
    const int cur_row = gwave + it * nwaves;
    if (cur_row >= nrows) break;
    const int nxt_row = cur_row + nwaves;
    const bool have_next = (nxt_row < nrows);     // wave-uniform
    const v4f* curbuf = (it & 1) ? buf1 : buf0;
    const v4f* nxtbuf = (it & 1) ? buf0 : buf1;

    // Prior ds reads of nxtbuf (from iteration it-1) must land before the
    // async engine overwrites it.
    asm volatile("s_wait_dscnt 0" ::: "memory");
    if (have_next)
      async_row_load(edata + (size_t)nxt_row * (LSTEPS * HID), nxtbuf, lane);

    // Async loads complete in order: cnt<=5 => the 5 loads of curbuf retired.
    if (have_next) asm volatile("s_wait_asynccnt 5" ::: "memory");
    else           asm volatile("s_wait_asynccnt 0" ::: "memory");

    v4f acc = (v4f)(0.0f);
#pragma unroll
    for (int l = 0; l < LSTEPS; ++l) {
      v4f e = curbuf[l * 32 + lane];   // ds_load_b128
      v4f r = rt[l];
      acc.x += e.x * r.x - e.y * r.y;
      acc.y += e.x * r.y + e.y * r.x;
      acc.z += e.z * r.z - e.w * r.w;
      acc.w += e.z * r.w + e.w * r.z;
    }
    acc *= 0.2f;                       // mean over L=5

    v4f* op = reinterpret_cast<v4f*>(out + (size_t)cur_row * HID) + lane;
    __builtin_nontemporal_store(acc, op);
  }
}

extern "C" void kernel_launch(void* const* d_in, const int* in_sizes, int n_in,
                              void* d_out, int out_size, void* d_ws, size_t ws_size,
                              hipStream_t stream) {
  const float* edata = (const float*)d_in[0];   // [N, 5, 128] fp32
  const float* rvec  = (const float*)d_in[1];   // [4, 64, 2] fp32
  float* out = (float*)d_out;                   // [N, 128] fp32

  const int nrows = in_sizes[0] / (LSTEPS * HID);

  int blocks = 2048;                            // 16384 waves -> 16 rows/wave at N=262144
  int max_blocks = (nrows + WAVES_PER_BLK - 1) / WAVES_PER_BLK;
  if (blocks > max_blocks) blocks = max_blocks;
  if (blocks < 1) blocks = 1;

  semenc_rotmean_kernel<<<dim3(blocks), dim3(THREADS), 0, stream>>>(edata, rvec, out, nrows);
}